// AVWGCN_2954937499758
// MI455X (gfx1250) — compile-verified
//
#include <hip/hip_runtime.h>
#include <math.h>

// Problem constants (from the reference)
#define NB    64     // B
#define NN    2048   // N
#define NCIN  64     // Cin
#define NCOUT 64     // Cout
#define NK    3      // Chebyshev order
#define ND    16     // embedding dim

typedef __attribute__((ext_vector_type(2))) float v2f;
typedef __attribute__((ext_vector_type(8))) float v8f;

// D = A(16x4) * B(4x16) + C, fp32 WMMA (CDNA5 V_WMMA_F32_16X16X4_F32)
__device__ __forceinline__ v8f wmma_f32(v2f a, v2f b, v8f c) {
    return __builtin_amdgcn_wmma_f32_16x16x4_f32(
        /*neg_a=*/false, a, /*neg_b=*/false, b,
        /*c_mod=*/(short)0, c, /*reuse_a=*/false, /*reuse_b=*/false);
}

// ---------------------------------------------------------------------------
// Kernel 1: A = softmax(relu(E @ E^T), axis=1).  One block per row.
// ---------------------------------------------------------------------------
__global__ void __launch_bounds__(256)
supports_kernel(const float* __restrict__ E, float* __restrict__ A) {
    const int n   = blockIdx.x;
    const int tid = threadIdx.x;
    __shared__ float en[ND];
    __shared__ float red[256];
    if (tid < ND) en[tid] = E[n * ND + tid];
    __syncthreads();

    float vals[8];
    float mx = 0.0f;                       // relu >= 0, so 0 is a valid floor
    #pragma unroll
    for (int j = 0; j < 8; ++j) {
        const int m = tid + j * 256;
        float dot = 0.0f;
        #pragma unroll
        for (int d = 0; d < ND; ++d) dot += en[d] * E[m * ND + d];
        float v = dot > 0.0f ? dot : 0.0f; // relu
        vals[j] = v;
        mx = fmaxf(mx, v);
    }
    red[tid] = mx; __syncthreads();
    for (int s = 128; s > 0; s >>= 1) {
        if (tid < s) red[tid] = fmaxf(red[tid], red[tid + s]);
        __syncthreads();
    }
    const float rmax = red[0];
    __syncthreads();
    float sum = 0.0f;
    #pragma unroll
    for (int j = 0; j < 8; ++j) { vals[j] = __expf(vals[j] - rmax); sum += vals[j]; }
    red[tid] = sum; __syncthreads();
    for (int s = 128; s > 0; s >>= 1) {
        if (tid < s) red[tid] += red[tid + s];
        __syncthreads();
    }
    const float inv = 1.0f / red[0];
    #pragma unroll
    for (int j = 0; j < 8; ++j) A[n * NN + tid + j * 256] = vals[j] * inv;
}

// ---------------------------------------------------------------------------
// Kernel 2: T2 = 2*(A@A) - I.  32x32 output per wave (2x2 WMMA tiles),
// 4 waves/block, 64*64 wave-tiles -> 1024 blocks.
// ---------------------------------------------------------------------------
__global__ void __launch_bounds__(128)
cheb2_kernel(const float* __restrict__ A, float* __restrict__ T2) {
    const int wave = threadIdx.x >> 5;
    const int lane = threadIdx.x & 31;
    const int g    = blockIdx.x * 4 + wave;
    const int tm   = g >> 6;                 // 0..63 (32-row tile)
    const int tn   = g & 63;                 // 0..63 (32-col tile)
    const int l16  = lane & 15;
    const int kb   = (lane >> 4) * 2;        // 0 or 2
    const int row0 = tm * 32 + l16;
    const int col0 = tn * 32 + l16;
    const long r0Base = (long)row0 * NN;
    const long r1Base = r0Base + 16L * NN;

    v8f acc00 = {}, acc01 = {}, acc10 = {}, acc11 = {};
    for (int kk = 0; kk < NN; kk += 4) {
        __builtin_prefetch(&A[r0Base + kk + 256], 0, 3);
        __builtin_prefetch(&A[r1Base + kk + 256], 0, 3);
        v2f a0, a1, b0, b1;
        a0.x = A[r0Base + kk + kb];     a0.y = A[r0Base + kk + kb + 1];
        a1.x = A[r1Base + kk + kb];     a1.y = A[r1Base + kk + kb + 1];
        const long kr0 = (long)(kk + kb) * NN;
        const long kr1 = kr0 + NN;
        b0.x = A[kr0 + col0];           b0.y = A[kr1 + col0];
        b1.x = A[kr0 + col0 + 16];      b1.y = A[kr1 + col0 + 16];
        acc00 = wmma_f32(a0, b0, acc00);
        acc01 = wmma_f32(a0, b1, acc01);
        acc10 = wmma_f32(a1, b0, acc10);
        acc11 = wmma_f32(a1, b1, acc11);
    }
    const int col1 = col0 + 16;
    #pragma unroll
    for (int r = 0; r < 8; ++r) {
        const int m0 = tm * 32 + r + 8 * (lane >> 4);
        const int m1 = m0 + 16;
        T2[(long)m0 * NN + col0] = 2.0f * acc00[r] - (m0 == col0 ? 1.0f : 0.0f);
        T2[(long)m0 * NN + col1] = 2.0f * acc01[r] - (m0 == col1 ? 1.0f : 0.0f);
        T2[(long)m1 * NN + col0] = 2.0f * acc10[r] - (m1 == col0 ? 1.0f : 0.0f);
        T2[(long)m1 * NN + col1] = 2.0f * acc11[r] - (m1 == col1 ? 1.0f : 0.0f);
    }
}

// ---------------------------------------------------------------------------
// Kernel 3: Y_k[b] = S_k @ X_b  (k=1: S=A, k=2: S=T2), output in [b][n][c].
// Each wave: 32 rows x 64 cols (2x4 WMMA tiles; A-frags reused 4x, the
// 8 B-loads per step cover one full contiguous 256B row of X).
// grid = (16, B, 2), block = 128 (4 waves).
// ---------------------------------------------------------------------------
__global__ void __launch_bounds__(128)
sgemm_sx_kernel(const float* __restrict__ S1, const float* __restrict__ S2,
                const float* __restrict__ X,
                float* __restrict__ Y1, float* __restrict__ Y2) {
    const float* __restrict__ S = (blockIdx.z == 0) ? S1 : S2;
    float*       __restrict__ Y = (blockIdx.z == 0) ? Y1 : Y2;
    const int b    = blockIdx.y;
    const int wave = threadIdx.x >> 5;
    const int lane = threadIdx.x & 31;
    const int tm   = blockIdx.x * 4 + wave;  // 0..63 (32-row tile)
    const int l16  = lane & 15;
    const int kb   = (lane >> 4) * 2;
    const int row0 = tm * 32 + l16;
    const long r0Base = (long)row0 * NN;
    const long r1Base = r0Base + 16L * NN;
    const float* __restrict__ Xb = X + (long)b * NN * NCIN;

    v8f acc[2][4] = {};
    for (int kk = 0; kk < NN; kk += 4) {
        __builtin_prefetch(&S[r0Base + kk + 256], 0, 3);
        __builtin_prefetch(&S[r1Base + kk + 256], 0, 3);
        v2f a0, a1;
        a0.x = S[r0Base + kk + kb];  a0.y = S[r0Base + kk + kb + 1];
        a1.x = S[r1Base + kk + kb];  a1.y = S[r1Base + kk + kb + 1];
        const long kr0 = (long)(kk + kb) * NCIN;
        const long kr1 = kr0 + NCIN;
        #pragma unroll
        for (int f = 0; f < 4; ++f) {
            v2f bb;
            bb.x = Xb[kr0 + f * 16 + l16];
            bb.y = Xb[kr1 + f * 16 + l16];
            acc[0][f] = wmma_f32(a0, bb, acc[0][f]);
            acc[1][f] = wmma_f32(a1, bb, acc[1][f]);
        }
    }
    float* __restrict__ Yb = Y + (long)b * NN * NCIN;
    #pragma unroll
    for (int f = 0; f < 4; ++f) {
        const int col = f * 16 + l16;
        #pragma unroll
        for (int r = 0; r < 8; ++r) {
            const int m0 = tm * 32 + r + 8 * (lane >> 4);
            Yb[(long)m0 * NCIN + col]        = acc[0][f][r];
            Yb[(long)(m0 + 16) * NCIN + col] = acc[1][f][r];
        }
    }
}

// ---------------------------------------------------------------------------
// Kernel 4: per-node output.  One block (256 thr = 8 waves) per node n.
//   W_n[ki][o] = sum_d E[n,d] * pool[d][k][i][o]   (LDS, 192x64)
//   G_n[b][ki] = XG_k[b][n][i]  (k=0->X, 1->Y1, 2->Y2; LDS, stride 198:
//                8B-aligned pairs + conflict-free banks for WMMA A loads)
//   out[b,n,o] = G_n @ W_n + bias_n.  Each wave: 16x32 (A-frag reused 2x).
// ---------------------------------------------------------------------------
#define GS_STRIDE 198
#define K4_SMEM_FLOATS (192 * 64 + 64 * GS_STRIDE + 64)

__global__ void __launch_bounds__(256)
node_out_kernel(const float* __restrict__ E, const float* __restrict__ wpool,
                const float* __restrict__ bpool,
                const float* __restrict__ X, const float* __restrict__ Y1,
                const float* __restrict__ Y2, float* __restrict__ out) {
    const int n   = blockIdx.x;
    const int tid = threadIdx.x;
    extern __shared__ float smem[];
    float* Ws    = smem;                       // 192*64
    float* Gs    = smem + 192 * 64;            // 64*198 (padded)
    float* biasS = Gs + 64 * GS_STRIDE;        // 64
    __shared__ float en[ND];

    if (tid < ND) en[tid] = E[n * ND + tid];
    __syncthreads();

    // Build W_n in LDS: Ws[ki*64 + o]
    for (int idx = tid; idx < 192 * 64; idx += 256) {
        const int ki = idx >> 6;               // 0..191
        const int o  = idx & 63;
        const int k  = ki >> 6;                // 0..2
        const int i  = ki & 63;
        float acc = 0.0f;
        #pragma unroll
        for (int d = 0; d < ND; ++d)
            acc += en[d] * wpool[(((long)d * NK + k) * NCIN + i) * NCOUT + o];
        Ws[idx] = acc;
    }
    if (tid < NCOUT) {
        float acc = 0.0f;
        #pragma unroll
        for (int d = 0; d < ND; ++d) acc += en[d] * bpool[d * NCOUT + tid];
        biasS[tid] = acc;
    }
    // Stage G_n in LDS: Gs[b*198 + ki]
    for (int idx = tid; idx < 64 * 192; idx += 256) {
        const int b  = idx / 192;
        const int ki = idx % 192;
        const int k  = ki >> 6;
        const int i  = ki & 63;
        const float* __restrict__ src = (k == 0) ? X : ((k == 1) ? Y1 : Y2);
        Gs[b * GS_STRIDE + ki] = src[((long)b * NN + n) * NCIN + i];
    }
    __syncthreads();

    const int wave = tid >> 5;                 // 0..7
    const int lane = tid & 31;
    const int tm   = wave >> 1;                // batch tile 0..3
    const int tnp  = (wave & 1) * 2;           // cout tile pair {0,1} or {2,3}
    const int l16  = lane & 15;
    const int kb   = (lane >> 4) * 2;
    const int arow = tm * 16 + l16;            // batch row within 64
    const int col0 = tnp * 16 + l16;

    v8f acc0 = {}, acc1 = {};
    #pragma unroll 4
    for (int kk = 0; kk < 192; kk += 4) {
        v2f a, b0, b1;
        a.x  = Gs[arow * GS_STRIDE + kk + kb];
        a.y  = Gs[arow * GS_STRIDE + kk + kb + 1];
        b0.x = Ws[(kk + kb)     * NCOUT + col0];
        b0.y = Ws[(kk + kb + 1) * NCOUT + col0];
        b1.x = Ws[(kk + kb)     * NCOUT + col0 + 16];
        b1.y = Ws[(kk + kb + 1) * NCOUT + col0 + 16];
        acc0 = wmma_f32(a, b0, acc0);
        acc1 = wmma_f32(a, b1, acc1);
    }
    const float bias0 = biasS[col0];
    const float bias1 = biasS[col0 + 16];
    #pragma unroll
    for (int r = 0; r < 8; ++r) {
        const int b = tm * 16 + r + 8 * (lane >> 4);
        out[((long)b * NN + n) * NCOUT + col0]      = acc0[r] + bias0;
        out[((long)b * NN + n) * NCOUT + col0 + 16] = acc1[r] + bias1;
    }
}

// ---------------------------------------------------------------------------
// Launcher.  Workspace layout (floats):
//   [0, N*N)                      A      (16 MB)
//   [N*N, 2*N*N)                  T2     (16 MB)
//   [2*N*N, +B*N*C)               Y1     (32 MB)
//   [.., +B*N*C)                  Y2     (32 MB)   total 96 MB
// ---------------------------------------------------------------------------
extern "C" void kernel_launch(void* const* d_in, const int* in_sizes, int n_in,
                              void* d_out, int out_size, void* d_ws, size_t ws_size,
                              hipStream_t stream) {
    const float* X  = (const float*)d_in[0];
    const float* E  = (const float*)d_in[1];
    const float* wp = (const float*)d_in[2];
    const float* bp = (const float*)d_in[3];
    float* out = (float*)d_out;

    float* ws = (float*)d_ws;
    float* A  = ws;
    float* T2 = ws + (long)NN * NN;
    float* Y1 = ws + 2L * NN * NN;
    float* Y2 = Y1 + (long)NB * NN * NCIN;

    // 1) adaptive adjacency
    supports_kernel<<<NN, 256, 0, stream>>>(E, A);
    // 2) T2 = 2 A A - I   (64x64 tiles of 32x32 per wave)
    cheb2_kernel<<<(64 * 64) / 4, 128, 0, stream>>>(A, T2);
    // 3) Y1 = A @ X_b, Y2 = T2 @ X_b for all batches
    sgemm_sx_kernel<<<dim3(16, NB, 2), 128, 0, stream>>>(A, T2, X, Y1, Y2);
    // 4) per-node dynamic-weight contraction + bias
    node_out_kernel<<<NN, 256, K4_SMEM_FLOATS * sizeof(float), stream>>>(
        E, wp, bp, X, Y1, Y2, out);
    (void)in_sizes; (void)n_in; (void)out_size; (void)ws_size;
}